// MangroveGCN_8143257993949
// MI455X (gfx1250) — compile-verified
//
#include <hip/hip_runtime.h>

#define N_NODES 100000
#define N_EDGES 3200000
#define N_GRAPHS 64
#define H 64

typedef __attribute__((ext_vector_type(2))) float v2f;
typedef __attribute__((ext_vector_type(8))) float v8f;

// ---------------------------------------------------------------------------
// WMMA helper: D = A(16x4 f32) * B(4x16 f32) + C, full fp32 (matches reference)
// ---------------------------------------------------------------------------
__device__ __forceinline__ v8f wmma4(v2f a, v2f b, v8f c) {
  return __builtin_amdgcn_wmma_f32_16x16x4_f32(false, a, false, b, (short)0, c,
                                               false, false);
}

// Wave computes a 16x16 tile: acc += A(16x64, LDS row-major) @ B(64x64 global)[:, n0:n0+16]
__device__ __forceinline__ void gemm_a_lds_b_glb(const float* __restrict__ As,
                                                 const float* __restrict__ Bg,
                                                 int n0, int lane, v8f& acc) {
  const int half = lane >> 4;
  const int l16  = lane & 15;
#pragma unroll
  for (int k = 0; k < H; k += 4) {
    const int ka = k + 2 * half;
    v2f a, b;
    a.x = As[l16 * H + ka];
    a.y = As[l16 * H + ka + 1];
    b.x = Bg[ka * H + n0 + l16];
    b.y = Bg[(ka + 1) * H + n0 + l16];
    acc = wmma4(a, b, acc);
  }
}

// ---------------------------------------------------------------------------
// Async copy global -> LDS (CDNA5 GLOBAL_LOAD_ASYNC_TO_LDS, ASYNCcnt-tracked).
// lds_off = low 32 bits of shared-aperture flat address == wave LDS offset.
// ---------------------------------------------------------------------------
__device__ __forceinline__ void async_copy_b128(unsigned lds_off, const void* g) {
  asm volatile("global_load_async_to_lds_b128 %0, %1, off"
               :
               : "v"(lds_off), "v"(g)
               : "memory");
}
__device__ __forceinline__ void wait_asynccnt0() {
  asm volatile("s_wait_asynccnt 0x0" ::: "memory");
}

// ---------------------------------------------------------------------------
// Encoder: h = LN(relu(relu(relu(x@W0+b0)@W1+b1)@W2+b2)) @ W3 + b3
// 16 nodes per 128-thread block, 4 waves x one 16-col N-tile each.
// ---------------------------------------------------------------------------
__global__ __launch_bounds__(128) void encoder_kernel(
    const float* __restrict__ x,
    const float* __restrict__ W0, const float* __restrict__ b0,
    const float* __restrict__ W1, const float* __restrict__ b1,
    const float* __restrict__ W2, const float* __restrict__ b2,
    const float* __restrict__ ge, const float* __restrict__ be,
    const float* __restrict__ W3, const float* __restrict__ b3,
    float* __restrict__ hout) {
  __shared__ float ldsX[16 * 9];
  __shared__ float A[16 * H];
  __shared__ float B[16 * H];
  const int tid  = threadIdx.x;
  const int base = blockIdx.x * 16;  // 100000 = 6250 * 16, all tiles full

  if (tid < 144) ldsX[tid] = x[base * 9 + tid];
  __syncthreads();

  // layer 0: 9 -> 64 (scalar; K=9 too small for WMMA)
#pragma unroll
  for (int i = 0; i < 8; ++i) {
    int idx = tid + 128 * i;
    int r = idx >> 6, c = idx & 63;
    float v = b0[c];
#pragma unroll
    for (int k = 0; k < 9; ++k) v += ldsX[r * 9 + k] * W0[k * H + c];
    A[idx] = fmaxf(v, 0.f);
  }
  __syncthreads();

  const int lane = tid & 31;
  const int wv   = tid >> 5;
  const int n0   = wv * 16;
  const int half = lane >> 4, l16 = lane & 15;

  {  // layer 1
    v8f acc;
    float bb = b1[n0 + l16];
#pragma unroll
    for (int v = 0; v < 8; ++v) acc[v] = bb;
    gemm_a_lds_b_glb(A, W1, n0, lane, acc);
#pragma unroll
    for (int v = 0; v < 8; ++v)
      B[(v + 8 * half) * H + n0 + l16] = fmaxf(acc[v], 0.f);
  }
  __syncthreads();
  {  // layer 2
    v8f acc;
    float bb = b2[n0 + l16];
#pragma unroll
    for (int v = 0; v < 8; ++v) acc[v] = bb;
    gemm_a_lds_b_glb(B, W2, n0, lane, acc);
#pragma unroll
    for (int v = 0; v < 8; ++v)
      A[(v + 8 * half) * H + n0 + l16] = fmaxf(acc[v], 0.f);
  }
  __syncthreads();

  // LayerNorm over H=64 per row (wave32: 2 values per lane, shfl_xor reduce)
#pragma unroll
  for (int rr = 0; rr < 4; ++rr) {
    int r = wv * 4 + rr;
    float v0 = A[r * H + 2 * lane], v1 = A[r * H + 2 * lane + 1];
    float s = v0 + v1, q = v0 * v0 + v1 * v1;
#pragma unroll
    for (int m = 16; m >= 1; m >>= 1) {
      s += __shfl_xor(s, m, 32);
      q += __shfl_xor(q, m, 32);
    }
    float mu  = s * (1.f / 64.f);
    float var = q * (1.f / 64.f) - mu * mu;
    float rs  = rsqrtf(var + 1e-5f);
    B[r * H + 2 * lane]     = (v0 - mu) * rs * ge[2 * lane] + be[2 * lane];
    B[r * H + 2 * lane + 1] = (v1 - mu) * rs * ge[2 * lane + 1] + be[2 * lane + 1];
  }
  __syncthreads();

  {  // layer 3 (no relu) -> global
    v8f acc;
    float bb = b3[n0 + l16];
#pragma unroll
    for (int v = 0; v < 8; ++v) acc[v] = bb;
    gemm_a_lds_b_glb(B, W3, n0, lane, acc);
#pragma unroll
    for (int v = 0; v < 8; ++v)
      hout[(long)(base + v + 8 * half) * H + n0 + l16] = acc[v];
  }
}

// ---------------------------------------------------------------------------
// Degree / reciprocal (computed once; degree is layer-invariant)
// ---------------------------------------------------------------------------
__global__ void degree_kernel(const int* __restrict__ ei, float* __restrict__ cnt) {
  int e = blockIdx.x * blockDim.x + threadIdx.x;
  if (e < N_EDGES) atomicAdd(&cnt[ei[N_EDGES + e]], 1.0f);
}
__global__ void invcnt_kernel(float* __restrict__ cnt) {
  int n = blockIdx.x * blockDim.x + threadIdx.x;
  if (n < N_NODES) cnt[n] = 1.0f / fmaxf(cnt[n], 1.0f);
}

// ---------------------------------------------------------------------------
// Edge scatter: one wave per edge. wid forced wave-uniform via readfirstlane
// so src/dst/inv_cnt become SMEM scalar loads (1 per wave, not 32 lane loads).
// Mean normalization folded in here: agg accumulates h[src] * (1/deg[dst]),
// so agg holds the mean directly. h and agg are L2-resident (<192 MB L2).
// ---------------------------------------------------------------------------
__global__ __launch_bounds__(256) void scatter_kernel(const int* __restrict__ ei,
                                                      const float* __restrict__ hin,
                                                      const float* __restrict__ inv_cnt,
                                                      float* __restrict__ agg) {
  int wid = __builtin_amdgcn_readfirstlane(
      (int)((blockIdx.x * blockDim.x + threadIdx.x) >> 5));
  int lane = threadIdx.x & 31;
  if (wid >= N_EDGES) return;
  int src  = ei[wid];             // uniform -> s_load
  int dst  = ei[N_EDGES + wid];   // uniform -> s_load
  float ic = inv_cnt[dst];        // uniform -> s_load
  const float2 v = *(const float2*)(hin + (long)src * H + lane * 2);
  atomicAdd(&agg[(long)dst * H + lane * 2], v.x * ic);
  atomicAdd(&agg[(long)dst * H + lane * 2 + 1], v.y * ic);
}

// ---------------------------------------------------------------------------
// SAGE update: h_out = relu(mean @ Wl + bl + h_in @ Wr). A-tiles (2 x 4 KB)
// staged into LDS with GLOBAL_LOAD_ASYNC_TO_LDS_B128 (ASYNCcnt), then two
// WMMA GEMM chains into one accumulator. 16 nodes / block, 4 waves.
// ---------------------------------------------------------------------------
__global__ __launch_bounds__(128) void sage_kernel(
    const float* __restrict__ hin, const float* __restrict__ mean,
    const float* __restrict__ Wl, const float* __restrict__ bl,
    const float* __restrict__ Wr, float* __restrict__ hout) {
  __shared__ float Am[16 * H];
  __shared__ float Ah[16 * H];
  const int tid  = threadIdx.x;
  const int base = blockIdx.x * 16;

  const unsigned amBase = (unsigned)(uintptr_t)(void*)Am;
  const unsigned ahBase = (unsigned)(uintptr_t)(void*)Ah;
  const float* gm = mean + (long)base * H;
  const float* gh = hin + (long)base * H;
#pragma unroll
  for (int i = 0; i < 2; ++i) {
    int q = (tid + 128 * i) * 4;  // float index; 4 floats per B128
    async_copy_b128(amBase + q * 4u, gm + q);
    async_copy_b128(ahBase + q * 4u, gh + q);
  }
  wait_asynccnt0();
  __syncthreads();

  const int lane = tid & 31;
  const int wv   = tid >> 5;
  const int n0   = wv * 16;
  const int half = lane >> 4, l16 = lane & 15;
  v8f acc;
  float bb = bl[n0 + l16];
#pragma unroll
  for (int v = 0; v < 8; ++v) acc[v] = bb;
  gemm_a_lds_b_glb(Am, Wl, n0, lane, acc);
  gemm_a_lds_b_glb(Ah, Wr, n0, lane, acc);
#pragma unroll
  for (int v = 0; v < 8; ++v)
    hout[(long)(base + v + 8 * half) * H + n0 + l16] = fmaxf(acc[v], 0.f);
}

// ---------------------------------------------------------------------------
// Pooling: batch is sorted, so run-length accumulate per thread; flush with
// atomicAdd + int-bits atomicMax (valid: post-ReLU values >= 0, init 0).
// Block covers 256 nodes; thread owns one column, stride-4 node walk.
// ---------------------------------------------------------------------------
__global__ __launch_bounds__(256) void pool_kernel(const float* __restrict__ h,
                                                   const int* __restrict__ batch,
                                                   float* __restrict__ psum,
                                                   float* __restrict__ pmax) {
  const int tid  = threadIdx.x;
  const int c    = tid & 63;
  const int q    = tid >> 6;  // 0..3
  const int base = blockIdx.x * 256;
  int gprev = -1;
  float s = 0.f, m = 0.f;
  for (int i = 0; i < 64; ++i) {
    int n = base + q + (i << 2);
    if (n >= N_NODES) break;
    int g = batch[n];
    if (g != gprev) {
      if (gprev >= 0) {
        atomicAdd(&psum[gprev * H + c], s);
        atomicMax((int*)&pmax[gprev * H + c], __float_as_int(m));
      }
      gprev = g;
      s = 0.f;
      m = 0.f;
    }
    float v = h[(long)n * H + c];
    s += v;
    m = fmaxf(m, v);
  }
  if (gprev >= 0) {
    atomicAdd(&psum[gprev * H + c], s);
    atomicMax((int*)&pmax[gprev * H + c], __float_as_int(m));
  }
}

// ---------------------------------------------------------------------------
// Head: relu([max|sum] @ Wlin + blin) -> LN -> @ Wf + bf ; tiny, one block.
// ---------------------------------------------------------------------------
__global__ __launch_bounds__(256) void head_kernel(
    const float* __restrict__ pmax, const float* __restrict__ psum,
    const float* __restrict__ Wlin, const float* __restrict__ blin,
    const float* __restrict__ gn, const float* __restrict__ bn,
    const float* __restrict__ Wf, const float* __restrict__ bf,
    float* __restrict__ out) {
  __shared__ float G[64 * 64];
  const int tid = threadIdx.x;
  for (int idx = tid; idx < 64 * 64; idx += 256) {
    int r = idx >> 6, c = idx & 63;
    float v = blin[c];
#pragma unroll 4
    for (int k = 0; k < 64; ++k) v += pmax[r * 64 + k] * Wlin[k * 64 + c];
#pragma unroll 4
    for (int k = 0; k < 64; ++k) v += psum[r * 64 + k] * Wlin[(64 + k) * 64 + c];
    G[idx] = fmaxf(v, 0.f);
  }
  __syncthreads();
  const int lane = tid & 31, wv = tid >> 5;
  for (int r = wv; r < 64; r += 8) {
    float v0 = G[r * 64 + 2 * lane], v1 = G[r * 64 + 2 * lane + 1];
    float s = v0 + v1, q = v0 * v0 + v1 * v1;
#pragma unroll
    for (int m = 16; m >= 1; m >>= 1) {
      s += __shfl_xor(s, m, 32);
      q += __shfl_xor(q, m, 32);
    }
    float mu = s * (1.f / 64.f), var = q * (1.f / 64.f) - mu * mu;
    float rs = rsqrtf(var + 1e-5f);
    G[r * 64 + 2 * lane]     = (v0 - mu) * rs * gn[2 * lane] + bn[2 * lane];
    G[r * 64 + 2 * lane + 1] = (v1 - mu) * rs * gn[2 * lane + 1] + bn[2 * lane + 1];
  }
  __syncthreads();
  if (tid < 128) {
    int r = tid >> 1, j = tid & 1;
    float v = bf[j];
#pragma unroll 4
    for (int k = 0; k < 64; ++k) v += G[r * 64 + k] * Wf[k * 2 + j];
    out[r * 2 + j] = v;
  }
}

// ---------------------------------------------------------------------------
// Launch: inputs are the flattened pytree of setup_inputs():
// 0:x 1:edge_index 2:batch 3..12:enc 13..27:sage(5x3) 28..33:head
// ---------------------------------------------------------------------------
extern "C" void kernel_launch(void* const* d_in, const int* in_sizes, int n_in,
                              void* d_out, int out_size, void* d_ws, size_t ws_size,
                              hipStream_t stream) {
  (void)in_sizes; (void)n_in; (void)out_size; (void)ws_size;
  const float* x    = (const float*)d_in[0];
  const int*   ei   = (const int*)d_in[1];
  const int*   batch= (const int*)d_in[2];
  const float* W0 = (const float*)d_in[3];
  const float* b0 = (const float*)d_in[4];
  const float* W1 = (const float*)d_in[5];
  const float* b1 = (const float*)d_in[6];
  const float* W2 = (const float*)d_in[7];
  const float* b2 = (const float*)d_in[8];
  const float* ge = (const float*)d_in[9];
  const float* be = (const float*)d_in[10];
  const float* W3 = (const float*)d_in[11];
  const float* b3 = (const float*)d_in[12];
  const float* Wlin = (const float*)d_in[28];
  const float* blin = (const float*)d_in[29];
  const float* gn   = (const float*)d_in[30];
  const float* bn   = (const float*)d_in[31];
  const float* Wf   = (const float*)d_in[32];
  const float* bf   = (const float*)d_in[33];

  float* ws   = (float*)d_ws;
  float* h0   = ws;                         // 6.4M f32
  float* h1   = ws + 6400000;               // 6.4M f32
  float* agg  = ws + 12800000;              // 6.4M f32 (holds mean directly)
  float* cnt  = ws + 19200000;              // 100k f32 (degree -> reciprocal)
  float* psum = ws + 19300096;              // 4096 f32
  float* pmax = ws + 19304192;              // 4096 f32

  // degree (once)
  hipMemsetAsync(cnt, 0, N_NODES * sizeof(float), stream);
  degree_kernel<<<(N_EDGES + 255) / 256, 256, 0, stream>>>(ei, cnt);
  invcnt_kernel<<<(N_NODES + 255) / 256, 256, 0, stream>>>(cnt);

  // encoder
  encoder_kernel<<<N_NODES / 16, 128, 0, stream>>>(x, W0, b0, W1, b1, W2, b2,
                                                   ge, be, W3, b3, h0);

  // 5 SAGE layers, ping-pong h buffers
  float* hin = h0;
  float* hout = h1;
  for (int l = 0; l < 5; ++l) {
    const float* Wl = (const float*)d_in[13 + 3 * l];
    const float* bl = (const float*)d_in[14 + 3 * l];
    const float* Wr = (const float*)d_in[15 + 3 * l];
    hipMemsetAsync(agg, 0, (size_t)N_NODES * H * sizeof(float), stream);
    scatter_kernel<<<N_EDGES / 8, 256, 0, stream>>>(ei, hin, cnt, agg);
    sage_kernel<<<N_NODES / 16, 128, 0, stream>>>(hin, agg, Wl, bl, Wr, hout);
    float* t = hin; hin = hout; hout = t;
  }

  // pooling + head
  hipMemsetAsync(psum, 0, 2 * N_GRAPHS * H * sizeof(float), stream);  // psum+pmax contiguous
  pool_kernel<<<(N_NODES + 255) / 256, 256, 0, stream>>>(hin, batch, psum, pmax);
  head_kernel<<<1, 256, 0, stream>>>(pmax, psum, Wlin, blin, gn, bn, Wf, bf,
                                     (float*)d_out);
}